// Calculator_61993557950977
// MI455X (gfx1250) — compile-verified
//
#include <hip/hip_runtime.h>
#include <hip/hip_bf16.h>

#define DIM 4096
#define NB  8192
#define NTHREADS 1024

typedef float v2f __attribute__((ext_vector_type(2)));
typedef float v8f __attribute__((ext_vector_type(8)));

struct f4 { float x, y, z, w; };

__device__ __forceinline__ f4 f4_add(f4 a, f4 b) {
    return f4{a.x + b.x, a.y + b.y, a.z + b.z, a.w + b.w};
}
__device__ __forceinline__ f4 f4_zero() { return f4{0.f, 0.f, 0.f, 0.f}; }

__device__ __forceinline__ f4 shfl_up4(f4 v, int delta) {
    f4 r;
    r.x = __shfl_up(v.x, delta, 32);
    r.y = __shfl_up(v.y, delta, 32);
    r.z = __shfl_up(v.z, delta, 32);
    r.w = __shfl_up(v.w, delta, 32);
    return r;
}

__global__ __launch_bounds__(NTHREADS) void calc_kernel(
    const float* __restrict__ betas,
    const float* __restrict__ lambdas,
    const float* __restrict__ gammas,
    float* __restrict__ out)
{
    // Prefix-sum table: S[j] = inclusive prefix over (g_j, g_j*d_j, g_j*log(lam_j), g_j*log1p(-lam_j))
    __shared__ f4 S[DIM];            // 64 KB; reused for thread partials in phase 3
    __shared__ f4 wtot[32];          // wave-total scan scratch
    __shared__ float P3[NTHREADS];   // 5th accumulator (n_I partials)

    const int t    = threadIdx.x;
    const int lane = t & 31;
    const int wave = t >> 5;

    __builtin_prefetch(betas + t, 0, 1);   // global_prefetch_b8 warm-up of betas

    // ---------------- Phase 1: per-element quantities + block scan ----------------
    // Thread t owns elements j = 4t .. 4t+3 (contiguous -> b128 global loads).
    const float4 g4 = reinterpret_cast<const float4*>(gammas)[t];
    const float4 l4 = reinterpret_cast<const float4*>(lambdas)[t];

    f4 q[4], incl[4];
    {
        float gs[4] = {g4.x, g4.y, g4.z, g4.w};
        float ls[4] = {l4.x, l4.y, l4.z, l4.w};
#pragma unroll
        for (int i = 0; i < 4; ++i) {
            const int j = 4 * t + i;
            const float kk   = (float)(j + 1);          // reference kk = 1..dim
            const float lamk = 1.0f - 1.0f / (kk + 1.0f);
            const float oml  = 1.0f - lamk;
            const float d    = logf(oml) - logf(lamk);  // log((1-lam_k)/lam_k)
            const float g    = gs[i];
            q[i].x = g;                                  // Gp element
            q[i].y = g * d;                              // Cp element
            q[i].z = g * logf(ls[i]);                    // Lp element
            q[i].w = g * log1pf(-ls[i]);                 // Mp element
        }
    }
    incl[0] = q[0];
#pragma unroll
    for (int i = 1; i < 4; ++i) incl[i] = f4_add(incl[i - 1], q[i]);
    const f4 tot = incl[3];

    // Intra-wave inclusive scan of thread totals (shuffle-based, wave32).
    f4 wscan = tot;
#pragma unroll
    for (int off = 1; off < 32; off <<= 1) {
        f4 up = shfl_up4(wscan, off);
        if (lane >= off) wscan = f4_add(wscan, up);
    }
    if (lane == 31) wtot[wave] = wscan;
    __syncthreads();

    if (wave == 0) {   // scan the 32 wave totals
        f4 w = wtot[lane];
#pragma unroll
        for (int off = 1; off < 32; off <<= 1) {
            f4 up = shfl_up4(w, off);
            if (lane >= off) w = f4_add(w, up);
        }
        wtot[lane] = w;   // inclusive scan of wave totals
    }
    __syncthreads();

    // Exclusive-before-this-thread = shfl_up(inclusive, 1)
    f4 excl = shfl_up4(wscan, 1);
    if (lane == 0) excl = f4_zero();
    f4 base = (wave > 0) ? wtot[wave - 1] : f4_zero();
    base = f4_add(base, excl);
#pragma unroll
    for (int i = 0; i < 4; ++i) S[4 * t + i] = f4_add(base, incl[i]);
    __syncthreads();

    // ---------------- Phase 2: per-beta gather + accumulate ----------------
    float s1 = 0.f, s2 = 0.f, s3 = 0.f, s4 = 0.f, s5 = 0.f;
#pragma unroll
    for (int i = 0; i < NB / NTHREADS; ++i) {
        const float beta  = betas[t + i * NTHREADS];
        const float lam   = 1.0f - 1.0f / beta;
        const float kfrac = 1.0f / (1.0f - lam) - 1.0f;
        const int   k     = (int)floorf(kfrac);          // in [1, DIM-1]
        const float kbf   = (float)k;
        const float lmax  = 1.0f - 1.0f / (kbf + 1.0f);
        const float A     = logf(lmax) - logf(1.0f - lmax);
        const f4 P = S[k - 1];                           // exclusive prefix at k (ds_load_b128)
        s1 += A * P.x;    // A_b * Gp
        s2 += P.y;        // Cp
        s3 += P.x;        // Gp      -> n_I
        s4 += P.z;        // Lp      -> S_L
        s5 += P.w;        // Mp      -> S_M
    }
    __syncthreads();
    S[t]  = f4{s1, s2, s4, s5};    // reuse prefix table as partials store
    P3[t] = s3;
    __syncthreads();

    // ---------------- Phase 3: wave-0 WMMA reduction of 1024 partials ----------------
    if (wave == 0) {
        // A-matrix layout for V_WMMA_F32_16X16X4_F32 (16x4, f32):
        //   lane L: M = L%16, K = {2*(L>=16), 2*(L>=16)+1} in the two A components.
        // Partition the 1024 thread-partials into 64 chunks of 16: chunk c = m*4 + k.
        const int m  = lane & 15;
        const int kh = lane >> 4;
        const int c0 = m * 4 + 2 * kh;
        const int c1 = c0 + 1;

        f4 a0 = f4_zero(), a1 = f4_zero();
        float a03 = 0.f, a13 = 0.f;
#pragma unroll
        for (int i = 0; i < 16; ++i) { a0 = f4_add(a0, S[c0 * 16 + i]); a03 += P3[c0 * 16 + i]; }
#pragma unroll
        for (int i = 0; i < 16; ++i) { a1 = f4_add(a1, S[c1 * 16 + i]); a13 += P3[c1 * 16 + i]; }

        const v2f ones = {1.0f, 1.0f};   // B = all-ones 4x16 (layout-independent)
        const v8f cz   = {};             // C = 0

        // D[m,n] = sum_k A[m,k]  -> row totals; total = sum over 16 rows.
        v2f A; v8f D;
        float r1, r2, r3, r4, r5;

        A = (v2f){a0.x, a1.x};
        D = __builtin_amdgcn_wmma_f32_16x16x4_f32(false, A, false, ones, (short)0, cz, false, false);
        r1 = D[0]+D[1]+D[2]+D[3]+D[4]+D[5]+D[6]+D[7];

        A = (v2f){a0.y, a1.y};
        D = __builtin_amdgcn_wmma_f32_16x16x4_f32(false, A, false, ones, (short)0, cz, false, false);
        r2 = D[0]+D[1]+D[2]+D[3]+D[4]+D[5]+D[6]+D[7];

        A = (v2f){a0.z, a1.z};
        D = __builtin_amdgcn_wmma_f32_16x16x4_f32(false, A, false, ones, (short)0, cz, false, false);
        r3 = D[0]+D[1]+D[2]+D[3]+D[4]+D[5]+D[6]+D[7];

        A = (v2f){a0.w, a1.w};
        D = __builtin_amdgcn_wmma_f32_16x16x4_f32(false, A, false, ones, (short)0, cz, false, false);
        r4 = D[0]+D[1]+D[2]+D[3]+D[4]+D[5]+D[6]+D[7];

        A = (v2f){a03, a13};
        D = __builtin_amdgcn_wmma_f32_16x16x4_f32(false, A, false, ones, (short)0, cz, false, false);
        r5 = D[0]+D[1]+D[2]+D[3]+D[4]+D[5]+D[6]+D[7];

        // Lanes 0-15 hold rows 0-7, lanes 16-31 hold rows 8-15: fold halves.
        r1 += __shfl_xor(r1, 16, 32);
        r2 += __shfl_xor(r2, 16, 32);
        r3 += __shfl_xor(r3, 16, 32);
        r4 += __shfl_xor(r4, 16, 32);
        r5 += __shfl_xor(r5, 16, 32);

        if (lane == 0) {
            const float ixt = r1 + r2;
            const float nI  = r5;
            const float gm_term      = expf(r3 / nI);          // prod(lam^mg)
            const float gm_comp_term = expf(r4 / nI);          // prod((1-lam)^mg)
            const float exp_term     = expf(2.0f * ixt / nI);
            const float log_term = -nI * 0.5f * logf(gm_comp_term + exp_term * gm_term);
            const float ity = ixt + log_term;
            const float rhs = 1.0f - ity / 1.0f;               // IXY = 1
            float l1 = 1.0f - ixt / 10.0f;                     // HX = 10
            if (l1 < 0.0f) l1 = fabsf(l1) * 20.0f;
            const float lhs = l1 * l1;                          // C=1, ALPHA=2
            out[0] = rhs;
            out[1] = lhs;
        }
    }
}

extern "C" void kernel_launch(void* const* d_in, const int* in_sizes, int n_in,
                              void* d_out, int out_size, void* d_ws, size_t ws_size,
                              hipStream_t stream) {
    const float* betas   = (const float*)d_in[0];
    const float* lambdas = (const float*)d_in[1];
    const float* gammas  = (const float*)d_in[2];
    float* out = (float*)d_out;
    (void)in_sizes; (void)n_in; (void)out_size; (void)d_ws; (void)ws_size;
    calc_kernel<<<1, NTHREADS, 0, stream>>>(betas, lambdas, gammas, out);
}